// DAS_22728966931062
// MI455X (gfx1250) — compile-verified
//
#include <hip/hip_runtime.h>
#include <math.h>

// DAS beamformer for MI455X (gfx1250).
// sensor_data: (8, 128, 16384) f32; sensor_mask: (8, 128, 2) f32
// out: (8, 256, 256) f32  (normalized image, transposed)

#define NB        8
#define NSENS     128
#define NT        16384          // samples per trace (64 KB fp32)
#define NPIX      65536          // 256*256 pixels per batch
#define NXDIM     256
#define TILE_PX   4096           // pixels per workgroup
#define PXPT      16             // pixels per thread (256 threads * 16)
#define CPY_RNDS  16             // 256 lanes * 16B * 16 rounds = 64 KB

typedef int v4i __attribute__((ext_vector_type(4)));
typedef __attribute__((address_space(1))) v4i gv4i_t;
typedef __attribute__((address_space(3))) v4i lv4i_t;

// Stage one 64KB sensor trace global -> LDS with async-to-LDS B128 copies.
// Each of the 256 threads issues CPY_RNDS b128 transfers (tracked on ASYNCcnt).
__device__ __forceinline__ void async_copy_trace(const float* __restrict__ gsrc,
                                                 float* lds_dst, int tid) {
  const float* gp = gsrc + tid * 4;   // 16 bytes per lane
  float*       lp = lds_dst + tid * 4;
#pragma unroll
  for (int r = 0; r < CPY_RNDS; ++r) {
    __builtin_amdgcn_global_load_async_to_lds_b128(
        (gv4i_t*)(gp + r * 1024), (lv4i_t*)(lp + r * 1024),
        /*imm offset*/ 0, /*cpol*/ 0);
  }
}

__global__ __launch_bounds__(256) void das_image_kernel(
    const float* __restrict__ sdata,   // (B, S, NT)
    const float* __restrict__ smask,   // (B, S, 2)
    float* __restrict__ img) {         // (B, NPIX) in x-major order
  __shared__ float trace[2][NT];       // 128 KB double buffer

  const int tid  = threadIdx.x;        // y coordinate (0..255)
  const int tile = blockIdx.x;         // 0..15 -> x rows [tile*16, tile*16+16)
  const int b    = blockIdx.y;         // batch

  const float DXf = 0.001f;
  const float DYf = 0.001f;
  const float DTf = (float)(1.0 / 12500000.0);

  float acc[PXPT];
#pragma unroll
  for (int k = 0; k < PXPT; ++k) acc[k] = 0.0f;

  const float* batch_data = sdata + (size_t)b * NSENS * NT;
  const float* batch_mask = smask + (size_t)b * NSENS * 2;

  // prologue: stage sensor 0 into buffer 0
  async_copy_trace(batch_data, trace[0], tid);

  const int iy0    = tid;               // zero-based pixel y
  const int ixbase = tile * PXPT;       // zero-based pixel x for k=0

  for (int s = 0; s < NSENS; ++s) {
    // overlap: kick off next trace DMA into the other buffer
    if (s + 1 < NSENS) {
      async_copy_trace(batch_data + (size_t)(s + 1) * NT, trace[(s + 1) & 1], tid);
      __builtin_amdgcn_s_wait_asynccnt(CPY_RNDS);  // current buffer's 16 done
    } else {
      __builtin_amdgcn_s_wait_asynccnt(0);
    }
    __syncthreads();  // all waves' async writes for buffer (s&1) are visible

    const float* tr = trace[s & 1];
    {
#pragma clang fp contract(off)
      // replicate reference op order exactly (no FMA contraction)
      float mxv = batch_mask[s * 2 + 0] * 1000.0f; mxv = mxv + 128.0f;
      float myv = batch_mask[s * 2 + 1] * 1000.0f; myv = myv + 128.0f;

      float wy  = myv - (float)(iy0 + 1); wy = wy + 1.0f;
      float fy  = wy * DYf;
      float fy2 = fy * fy;

#pragma unroll
      for (int k = 0; k < PXPT; ++k) {
        float wx  = mxv - (float)(ixbase + k + 1); wx = wx + 1.0f;
        float fx  = wx * DXf;
        float dis = __builtin_sqrtf(fx * fx + fy2);
        float tf  = dis / 1500.0f;
        tf        = tf / DTf;
        int t = (int)tf;                     // trunc, dis >= 0
        t = t < 0 ? 0 : (t > NT - 1 ? NT - 1 : t);
        acc[k] += tr[t];                     // ds_load_b32 gather
      }
    }
    __syncthreads();  // everyone done reading buffer (s&1) before reuse
  }

  // write image: linear p = x*256 + y, tile covers 4096 consecutive p
  float* op = img + (size_t)b * NPIX + (size_t)tile * TILE_PX;
#pragma unroll
  for (int k = 0; k < PXPT; ++k)
    op[k * NXDIM + tid] = acc[k];
}

__global__ __launch_bounds__(256) void das_minmax_kernel(
    const float* __restrict__ img, float* __restrict__ mm) {
  const int b   = blockIdx.x;
  const int tid = threadIdx.x;
  const float* p = img + (size_t)b * NPIX;
  float mn =  INFINITY;
  float mx = -INFINITY;
  for (int i = tid; i < NPIX; i += 256) {
    float v = p[i];
    mn = fminf(mn, v);
    mx = fmaxf(mx, v);
  }
  __shared__ float smn[256];
  __shared__ float smx[256];
  smn[tid] = mn;
  smx[tid] = mx;
  __syncthreads();
  for (int off = 128; off > 0; off >>= 1) {
    if (tid < off) {
      smn[tid] = fminf(smn[tid], smn[tid + off]);
      smx[tid] = fmaxf(smx[tid], smx[tid + off]);
    }
    __syncthreads();
  }
  if (tid == 0) {
    mm[b * 2 + 0] = smn[0];
    mm[b * 2 + 1] = smx[0];
  }
}

__global__ __launch_bounds__(256) void das_norm_kernel(
    const float* __restrict__ img, const float* __restrict__ mm,
    float* __restrict__ out) {
  const int gid = blockIdx.x * 256 + threadIdx.x;   // 0 .. 8*65536-1
  const int b   = gid >> 16;
  const int r   = gid & (NPIX - 1);
  const int y   = r >> 8;    // output row
  const int x   = r & 255;   // output col
  const float mn = mm[b * 2 + 0];
  const float mx = mm[b * 2 + 1];
  const float v  = img[(size_t)b * NPIX + x * NXDIM + y];  // transpose read
  out[gid] = (v - mn) / (mx - mn);
}

extern "C" void kernel_launch(void* const* d_in, const int* in_sizes, int n_in,
                              void* d_out, int out_size, void* d_ws, size_t ws_size,
                              hipStream_t stream) {
  (void)in_sizes; (void)n_in; (void)out_size; (void)ws_size;
  const float* sdata = (const float*)d_in[0];   // (8,128,16384)
  const float* smask = (const float*)d_in[1];   // (8,128,2)
  float* out = (float*)d_out;                   // (8,256,256)

  float* img = (float*)d_ws;                    // 8*65536 floats = 2 MB
  float* mm  = img + (size_t)NB * NPIX;         // 16 floats

  dim3 grid1(NPIX / TILE_PX, NB);               // (16, 8)
  das_image_kernel<<<grid1, 256, 0, stream>>>(sdata, smask, img);
  das_minmax_kernel<<<NB, 256, 0, stream>>>(img, mm);
  das_norm_kernel<<<(NB * NPIX) / 256, 256, 0, stream>>>(img, mm, out);
}